// HashGrid2D_11269994184713
// MI455X (gfx1250) — compile-verified
//
#include <hip/hip_runtime.h>
#include <stdint.h>

// HashGrid2D gather for gfx1250 (MI455X).
// positions: [N,2] f32 in [0, 4096), table: [2^19, 8] f32, out: [N, 8] f32.
// Memory-bound: 320 MB HBM-side traffic -> ~13.7 us floor @ 23.3 TB/s.
// Table (16 MB) stays L2-resident; pos/out streams use TH_NT hints.
//
// Hash note: reference uses int64 wraparound + arithmetic shifts. For
// ix,iy in [0, 2^12) every intermediate is nonnegative and < 2^45, so
// logical-shift uint64 arithmetic is bit-exact vs. the int64 reference.
// This avoids the compiler's ~20-VALU generic f32->i64 conversion pair.

#define HASH_BITS 19
#define HASH_MASK ((1u << HASH_BITS) - 1u)

typedef float v2f __attribute__((ext_vector_type(2)));
typedef float v4f __attribute__((ext_vector_type(4)));

__global__ __launch_bounds__(256) void hashgrid2d_gather_kernel(
    const v2f* __restrict__ pos,    // [N] packed (x,y)
    const v4f* __restrict__ table,  // [HASH_SIZE * 2] (each row = 2 x float4)
    v4f* __restrict__ out,          // [N * 2]
    int n)
{
    const int i = blockIdx.x * blockDim.x + threadIdx.x;
    if (i >= n) return;

    // Streaming read-once: non-temporal, keep L2 for the table.
    const v2f p = __builtin_nontemporal_load(&pos[i]);

    // CELL_SIZE == 1.0; coords nonneg => floor == trunc after v_floor_f32.
    const uint32_t ix = (uint32_t)floorf(p.x);   // v_floor_f32 + v_cvt_u32_f32
    const uint32_t iy = (uint32_t)floorf(p.y);

    // Murmur-style bitmix (bit-exact vs int64 reference for in-range coords).
    const uint32_t a = ix ^ (ix >> 16);
    uint64_t h = (uint64_t)a * 2246822507ull;    // v_mad_u64_u32
    h ^= (h >> 13);
    h += (uint64_t)iy * 3266489909ull;           // v_mad_u64_u32 (+ carry add)
    // Only low 19 bits of (h ^ (h >> 16)) are needed: do it in 32-bit.
    const uint32_t idx = ((uint32_t)h ^ (uint32_t)(h >> 16)) & HASH_MASK;

    // Gather one 32B row (2 x b128). Default temporal policy: hot in L2.
    const v4f* row = table + ((size_t)idx << 1);
    const v4f r0 = row[0];
    const v4f r1 = row[1];

    // Write-once output stream: non-temporal b128 stores.
    v4f* o = out + ((size_t)i << 1);
    __builtin_nontemporal_store(r0, &o[0]);
    __builtin_nontemporal_store(r1, &o[1]);
}

extern "C" void kernel_launch(void* const* d_in, const int* in_sizes, int n_in,
                              void* d_out, int out_size, void* d_ws, size_t ws_size,
                              hipStream_t stream) {
    (void)n_in; (void)out_size; (void)d_ws; (void)ws_size;

    const v2f* pos   = (const v2f*)d_in[0];   // positions: in_sizes[0] = 2*N floats
    const v4f* table = (const v4f*)d_in[1];   // table: HASH_SIZE * 8 floats
    v4f*       out   = (v4f*)d_out;           // N * 8 floats

    const int n = in_sizes[0] / 2;            // number of positions
    const int block = 256;                    // 8 wave32 waves per workgroup
    const int grid  = (n + block - 1) / block;

    hashgrid2d_gather_kernel<<<grid, block, 0, stream>>>(pos, table, out, n);
}